// RoIHeads_87557203296330
// MI455X (gfx1250) — compile-verified
//
#include <hip/hip_runtime.h>
#include <math.h>

#define DEVFN __device__ __forceinline__

// ---------------- problem constants (match reference) ----------------
constexpr int   B_   = 4, N_ = 4000, G_ = 64, C_ = 91;
constexpr int   NP_  = N_ + G_;            // 4064
constexpr int   NC_  = N_ * (C_ - 1);      // 360000 fg candidates / image
constexpr int   TOPK_ = 2048;
constexpr int   DET_  = 100;
constexpr float WF_ = 1333.0f, HF_ = 800.0f;      // x clipped to [0,W], y to [0,H]
constexpr float XCLIP_   = 4.135166556742356f;    // log(1000/16)
constexpr float SCORE_T_ = 0.05f, NMS_T_ = 0.5f, MIN_SZ_ = 0.01f;
constexpr float LBL_OFF_ = 1334.0f;               // max(W,H)+1
constexpr int   NBUCKET_ = 4096;
constexpr int   BCAP_    = 6144;                  // boundary-bucket capacity
constexpr int   SORTN_   = 8192;

// output layout (floats, concatenated in reference return order)
constexpr size_t OFF_DET_  = 0;                                  // [B,100,5]
constexpr size_t OFF_REG_  = (size_t)B_ * DET_ * 5;              // [B,4064,4]
constexpr size_t OFF_LBL_  = OFF_REG_ + (size_t)B_ * NP_ * 4;    // [B,4064]
constexpr size_t OFF_DETL_ = OFF_LBL_ + (size_t)B_ * NP_;        // [B,100]

typedef __attribute__((ext_vector_type(2))) float v2f;
typedef __attribute__((ext_vector_type(8))) float v8f;

#if defined(__gfx1250__) && __has_builtin(__builtin_amdgcn_wmma_f32_16x16x4_f32)
#define USE_WMMA_ROWSUM 1
#else
#define USE_WMMA_ROWSUM 0
#endif

// ---------------- CDNA5 async copy helpers (ASYNCcnt path) ----------------
DEVFN void async_copy16_to_lds(const void* gptr, void* lptr) {
#if defined(__gfx1250__)
  unsigned lds = (unsigned)(unsigned long long)lptr;   // addr[31:0] == LDS offset
  asm volatile("global_load_async_to_lds_b128 %0, %1, off"
               :: "v"(lds), "v"(gptr) : "memory");
#else
  (void)gptr; (void)lptr;
#endif
}
DEVFN void wait_async0() {
#if defined(__gfx1250__)
  asm volatile("s_wait_asynccnt 0" ::: "memory");
#endif
}

// ---------------- shared decode helper ----------------
DEVFN float4 decode_clip(const float* __restrict__ proposals,
                         const float* __restrict__ regr,
                         int b, int row, int col) {
  const float* pr = proposals + ((size_t)b * N_ + row) * 4;
  float px1 = pr[0], py1 = pr[1], px2 = pr[2], py2 = pr[3];
  float pw = px2 - px1, ph = py2 - py1;
  float pcx = px1 + 0.5f * pw, pcy = py1 + 0.5f * ph;
  const float* d = regr + (((size_t)b * N_ + row) * C_ + col) * 4;
  float dx = d[0] * 0.1f, dy = d[1] * 0.1f;
  float dw = fminf(d[2] * 0.2f, XCLIP_), dh = fminf(d[3] * 0.2f, XCLIP_);
  float cx = dx * pw + pcx, cy = dy * ph + pcy;
  float w = expf(dw) * pw, hh = expf(dh) * ph;
  float4 r;
  r.x = fminf(fmaxf(cx - 0.5f * w,  0.f), WF_);
  r.y = fminf(fmaxf(cy - 0.5f * hh, 0.f), HF_);
  r.z = fminf(fmaxf(cx + 0.5f * w,  0.f), WF_);
  r.w = fminf(fmaxf(cy + 0.5f * hh, 0.f), HF_);
  return r;
}

// ---------------- K0: zero scratch + det output regions ----------------
__global__ void k0_init(unsigned* hist, unsigned* selA_cnt, unsigned* selB_cnt,
                        unsigned* keepInit, float* out) {
  int i = blockIdx.x * blockDim.x + threadIdx.x;
  if (i < B_ * NBUCKET_) hist[i] = 0u;
  if (i < B_) { selA_cnt[i] = 0u; selB_cnt[i] = 0u; }
  if (i < B_ * 64) keepInit[i] = 0u;
  if (i < B_ * DET_ * 5) out[OFF_DET_ + i] = 0.f;
  if (i < B_ * DET_)     out[OFF_DETL_ + i] = 0.f;
}

// ---------------- K1: proposal<->gt matching, labels, box encoding ----------------
__global__ __launch_bounds__(256) void k1_match(const float* __restrict__ proposals,
                                                const float* __restrict__ gtb,
                                                const int* __restrict__ gtl,
                                                float* __restrict__ out) {
  __shared__ float4 g[G_];
  __shared__ float  garea[G_];
  __shared__ int    gl[G_];
  const int b = blockIdx.x;
  const int j = blockIdx.y * 256 + threadIdx.x;
  if (threadIdx.x < G_) {
    const float* p = gtb + ((size_t)b * G_ + threadIdx.x) * 4;
    float4 bb = make_float4(p[0], p[1], p[2], p[3]);
    g[threadIdx.x] = bb;
    garea[threadIdx.x] = (bb.z - bb.x) * (bb.w - bb.y);
    gl[threadIdx.x] = gtl[b * G_ + threadIdx.x];
  }
  __syncthreads();
  if (j >= NP_) return;
  float4 p;
  if (j < N_) {
    const float* pp = proposals + ((size_t)b * N_ + j) * 4;
    p = make_float4(pp[0], pp[1], pp[2], pp[3]);
  } else {
    p = g[j - N_];
  }
  float areaP = (p.z - p.x) * (p.w - p.y);
  float best = -1.f; int bi = 0;
  #pragma unroll 8
  for (int q = 0; q < G_; ++q) {
    float4 a = g[q];
    float iw = fminf(a.z, p.z) - fmaxf(a.x, p.x);
    float ih = fminf(a.w, p.w) - fmaxf(a.y, p.y);
    float inter = fmaxf(iw, 0.f) * fmaxf(ih, 0.f);
    float iou = inter / (garea[q] + areaP - inter);
    if (iou > best) { best = iou; bi = q; }   // first max wins (argmax semantics)
  }
  out[OFF_LBL_ + (size_t)b * NP_ + j] = (best >= 0.5f) ? (float)gl[bi] : 0.f;
  float4 gt = g[bi];
  float pw = p.z - p.x, ph = p.w - p.y;
  float pcx = p.x + 0.5f * pw, pcy = p.y + 0.5f * ph;
  float gw = gt.z - gt.x, gh = gt.w - gt.y;
  float gcx = gt.x + 0.5f * gw, gcy = gt.y + 0.5f * gh;
  float* rt = out + OFF_REG_ + ((size_t)b * NP_ + j) * 4;
  rt[0] = 10.f * (gcx - pcx) / pw;
  rt[1] = 10.f * (gcy - pcy) / ph;
  rt[2] = 5.f * logf(gw / pw);
  rt[3] = 5.f * logf(gh / ph);
}

// ---------------- K2: decode + softmax + filter + score histogram ----------------
// One wave per 16 proposal rows. Row-sum of 91 exps done with WMMA f32 16x16x4:
// A = ones (layout-independent), B columns (N = lane%16) carry the 16 rows' exps.
__global__ __launch_bounds__(256) void k2_scores(const float* __restrict__ proposals,
                                                 const float* __restrict__ logits,
                                                 const float* __restrict__ regr,
                                                 float* __restrict__ sm,
                                                 unsigned* __restrict__ hist) {
  const int b    = blockIdx.x;
  const int wave = threadIdx.x >> 5;
  const int lane = threadIdx.x & 31;
  const int tile = blockIdx.y * 8 + wave;
  if (tile >= N_ / 16) return;                 // wave-uniform
  const int h = lane >> 4;                     // 0/1: which half of each 4-col chunk
  const int r = lane & 15;                     // data row within tile == WMMA column
  const int row = tile * 16 + r;
  const float* lg = logits + ((size_t)b * N_ + row) * C_;

  // row max (stable softmax): lanes r and r+16 split the 91 columns
  float mx = -1e30f;
  for (int c = 0; c < 23; ++c) {
    int col = 4 * c + 2 * h;
    float l0 = lg[col];                                    // col <= 90 always
    float l1 = lg[min(col + 1, C_ - 1)];
    mx = fmaxf(mx, l0);
    mx = fmaxf(mx, (col + 1 < C_) ? l1 : -1e30f);
  }
  mx = fmaxf(mx, __shfl_xor(mx, 16, 32));

  // softmax denominator
  float denom;
#if USE_WMMA_ROWSUM
  v8f acc = {};
  v2f ones = {1.0f, 1.0f};
  for (int c = 0; c < 23; ++c) {
    int col = 4 * c + 2 * h;
    float e0 = expf(lg[col] - mx);
    float e1 = expf(lg[min(col + 1, C_ - 1)] - mx);
    v2f bv;
    bv.x = e0;
    bv.y = (col + 1 < C_) ? e1 : 0.f;
    acc = __builtin_amdgcn_wmma_f32_16x16x4_f32(
        false, ones, false, bv, (short)0, acc, false, false);
  }
  denom = acc[0];                 // D[.,N=lane%16] = rowsum(row) in every lane
#else
  float part = 0.f;
  for (int c = 0; c < 23; ++c) {
    int col = 4 * c + 2 * h;
    part += expf(lg[col] - mx);
    float e1 = expf(lg[min(col + 1, C_ - 1)] - mx);
    part += (col + 1 < C_) ? e1 : 0.f;
  }
  denom = part + __shfl_xor(part, 16, 32);
#endif
  const float inv = 1.0f / denom;

  // score + validity + histogram; store masked score
  const float* pr = proposals + ((size_t)b * N_ + row) * 4;
  float px1 = pr[0], py1 = pr[1], px2 = pr[2], py2 = pr[3];
  float pw = px2 - px1, phh = py2 - py1;
  float pcx = px1 + 0.5f * pw, pcy = py1 + 0.5f * phh;
  const float* rg = regr + ((size_t)b * N_ + row) * (size_t)(C_ * 4);
  float* smr = sm + (size_t)b * NC_ + (size_t)row * (C_ - 1);
  unsigned* hb = hist + b * NBUCKET_;
  for (int c = 0; c < 23; ++c) {
    for (int t = 0; t < 2; ++t) {
      int col = 4 * c + 2 * h + t;
      if (col >= 1 && col < C_) {
        float s = expf(lg[col] - mx) * inv;
        const float4 d = *(const float4*)(rg + col * 4);
        float dx = d.x * 0.1f, dy = d.y * 0.1f;
        float dw = fminf(d.z * 0.2f, XCLIP_), dh = fminf(d.w * 0.2f, XCLIP_);
        float cx = dx * pw + pcx, cy = dy * phh + pcy;
        float w = expf(dw) * pw, hh = expf(dh) * phh;
        float x1 = fminf(fmaxf(cx - 0.5f * w,  0.f), WF_);
        float y1 = fminf(fmaxf(cy - 0.5f * hh, 0.f), HF_);
        float x2 = fminf(fmaxf(cx + 0.5f * w,  0.f), WF_);
        float y2 = fminf(fmaxf(cy + 0.5f * hh, 0.f), HF_);
        bool valid = (s > SCORE_T_) && (x2 - x1 >= MIN_SZ_) && (y2 - y1 >= MIN_SZ_);
        smr[col - 1] = valid ? s : -1.0f;
        if (valid) atomicAdd(&hb[min(NBUCKET_ - 1, (int)(s * (float)NBUCKET_))], 1u);
      }
    }
  }
}

// ---------------- K3: find top-2048 threshold bucket per image ----------------
__global__ void k3_thresh(const unsigned* __restrict__ hist, int* __restrict__ tstar) {
  int b = threadIdx.x;
  if (b >= B_) return;
  const unsigned* hb = hist + b * NBUCKET_;
  unsigned cum = 0; int t = -1;
  for (int u = NBUCKET_ - 1; u >= 0; --u) {
    cum += hb[u];
    if (cum >= (unsigned)TOPK_) { t = u; break; }
  }
  tstar[b] = t;     // -1 => fewer than 2048 valid candidates: take all
}

// ---------------- K4: compact candidates above / at threshold bucket ----------------
__global__ __launch_bounds__(256) void k4_compact(const float* __restrict__ sm,
                                                  const int* __restrict__ tstar,
                                                  unsigned* __restrict__ selA_cnt,
                                                  unsigned* __restrict__ selB_cnt,
                                                  unsigned long long* __restrict__ selA,
                                                  unsigned long long* __restrict__ selB) {
  const int b = blockIdx.x;
  const int i = blockIdx.y * 256 + threadIdx.x;
  if (i >= NC_) return;
  float s = sm[(size_t)b * NC_ + i];
  if (s <= 0.f) return;
  int bucket = min(NBUCKET_ - 1, (int)(s * (float)NBUCKET_));
  int t = tstar[b];
  unsigned long long key =
      ((unsigned long long)__float_as_uint(s) << 32) | (unsigned)(~(unsigned)i);
  if (bucket > t) {
    unsigned p = atomicAdd(&selA_cnt[b], 1u);
    if (p < (unsigned)TOPK_) selA[(size_t)b * TOPK_ + p] = key;
  } else if (bucket == t) {
    unsigned p = atomicAdd(&selB_cnt[b], 1u);
    if (p < (unsigned)BCAP_) selB[(size_t)b * BCAP_ + p] = key;
  }
}

// ---------------- K5: LDS bitonic sort (desc) -> exact top-2048 + NMS prep ----------------
__global__ __launch_bounds__(1024) void k5_sort(const unsigned* __restrict__ selA_cnt,
                                                const unsigned* __restrict__ selB_cnt,
                                                const unsigned long long* __restrict__ selA,
                                                const unsigned long long* __restrict__ selB,
                                                const float* __restrict__ proposals,
                                                const float* __restrict__ regr,
                                                float* __restrict__ cb,
                                                float* __restrict__ ob,
                                                int* __restrict__ cl,
                                                float* __restrict__ cs,
                                                unsigned* __restrict__ keepInit) {
  __shared__ unsigned long long keys[SORTN_];      // 64 KB LDS (WGP has 320 KB)
  const int b = blockIdx.x, tid = threadIdx.x;
  const int nA = min((int)selA_cnt[b], TOPK_);
  const int nB = min((int)selB_cnt[b], BCAP_);
  for (int i = tid; i < SORTN_; i += 1024) {
    unsigned long long k = 0ull;
    if (i < nA) k = selA[(size_t)b * TOPK_ + i];
    else if (i - nA < nB) k = selB[(size_t)b * BCAP_ + (i - nA)];
    keys[i] = k;
  }
  __syncthreads();
  for (int k = 2; k <= SORTN_; k <<= 1) {
    for (int j = k >> 1; j > 0; j >>= 1) {
      for (int t = tid; t < SORTN_; t += 1024) {
        int ixj = t ^ j;
        if (ixj > t) {
          bool desc = ((t & k) == 0);
          unsigned long long a = keys[t], c = keys[ixj];
          bool sw = desc ? (a < c) : (a > c);
          if (sw) { keys[t] = c; keys[ixj] = a; }
        }
      }
      __syncthreads();
    }
  }
  // emit the exact top-2048 (desc by score, ties by ascending flat index)
  for (int i = tid; i < TOPK_; i += 1024) {
    unsigned long long key = keys[i];
    bool val = (key != 0ull);
    float4 bx = make_float4(0.f, 0.f, 0.f, 0.f);
    int label = 0; float sc = 0.f;
    if (val) {
      unsigned flat = ~(unsigned)(key & 0xffffffffull);
      sc = __uint_as_float((unsigned)(key >> 32));
      int row = flat / (C_ - 1);
      int col = flat % (C_ - 1) + 1;
      bx = decode_clip(proposals, regr, b, row, col);
      label = col;
      atomicOr(&keepInit[b * 64 + (i >> 5)], 1u << (i & 31));
    }
    float* cbp = cb + ((size_t)b * TOPK_ + i) * 4;
    cbp[0] = bx.x; cbp[1] = bx.y; cbp[2] = bx.z; cbp[3] = bx.w;
    float* obp = ob + ((size_t)b * TOPK_ + i) * 4;
    if (val) {
      float off = (float)label * LBL_OFF_;
      obp[0] = bx.x + off; obp[1] = bx.y + off; obp[2] = bx.z + off; obp[3] = bx.w + off;
    } else {
      obp[0] = -1e6f; obp[1] = -1e6f; obp[2] = -1e6f + 1.f; obp[3] = -1e6f + 1.f;
    }
    cl[b * TOPK_ + i] = label;
    cs[b * TOPK_ + i] = sc;
  }
}

// ---------------- K6: NMS suppression bitmask (boxes staged via async->LDS) ----------------
__global__ __launch_bounds__(256) void k6_mask(const float* __restrict__ ob,
                                               unsigned* __restrict__ mask) {
  __shared__ float4 lob[TOPK_];                 // 32 KB
  const int b = blockIdx.x;
  const float4* src = (const float4*)(ob + (size_t)b * TOPK_ * 4);
  #pragma unroll
  for (int k = 0; k < TOPK_ / 256; ++k) {
    int idx = threadIdx.x + k * 256;
    async_copy16_to_lds(src + idx, &lob[idx]);  // ASYNCcnt path
  }
  wait_async0();
  __syncthreads();
  const int base = blockIdx.y * 8;              // 8 rows per block
  for (int t = threadIdx.x; t < 8 * 64; t += 256) {
    int ri = base + (t >> 6);
    int w  = t & 63;
    float4 a = lob[ri];
    float areaA = (a.z - a.x) * (a.w - a.y);
    unsigned bits = 0u;
    int j0 = w * 32;
    #pragma unroll 8
    for (int jj = 0; jj < 32; ++jj) {
      int j = j0 + jj;
      if (j > ri) {
        float4 c = lob[j];
        float iw = fminf(a.z, c.z) - fmaxf(a.x, c.x);
        float ih = fminf(a.w, c.w) - fmaxf(a.y, c.y);
        float inter = fmaxf(iw, 0.f) * fmaxf(ih, 0.f);
        float areaB = (c.z - c.x) * (c.w - c.y);
        float iou = inter / (areaA + areaB - inter);
        if (iou > NMS_T_) bits |= (1u << jj);
      }
    }
    mask[((size_t)b * TOPK_ + ri) * 64 + w] = bits;
  }
}

// ---------------- K7: single-wave serial scan, early exit at 100 kept ----------------
__global__ __launch_bounds__(32) void k7_scan(const unsigned* __restrict__ mask,
                                              const unsigned* __restrict__ keepInit,
                                              const float* __restrict__ cb,
                                              const int* __restrict__ cl,
                                              const float* __restrict__ cs,
                                              float* __restrict__ out) {
  const int b = blockIdx.x, lane = threadIdx.x;
  const unsigned* m = mask + (size_t)b * TOPK_ * 64;
  unsigned k0 = keepInit[b * 64 + 2 * lane];
  unsigned k1 = keepInit[b * 64 + 2 * lane + 1];
  int kept = 0;
  for (int i = 0; i < TOPK_ && kept < DET_; ++i) {
    const unsigned* row = m + (size_t)i * 64;
    __builtin_prefetch(row + 64, 0, 0);         // global_prefetch_b8: next row
    __builtin_prefetch(row + 96, 0, 0);
    unsigned w0 = row[2 * lane];
    unsigned w1 = row[2 * lane + 1];
    unsigned myword = ((i >> 5) & 1) ? k1 : k0;
    unsigned kw = __shfl(myword, i >> 6, 32);   // broadcast owner's keep word
    if ((kw >> (i & 31)) & 1u) {                // wave-uniform
      k0 &= ~w0; k1 &= ~w1;                     // row i suppresses j>i
      if (lane == 0) {
        const float* bx = cb + ((size_t)b * TOPK_ + i) * 4;
        float* dr = out + OFF_DET_ + ((size_t)b * DET_ + kept) * 5;
        dr[0] = bx[0]; dr[1] = bx[1]; dr[2] = bx[2]; dr[3] = bx[3];
        dr[4] = cs[b * TOPK_ + i];
        out[OFF_DETL_ + (size_t)b * DET_ + kept] = (float)cl[b * TOPK_ + i];
      }
      ++kept;
    }
  }
}

// ---------------- host launch ----------------
extern "C" void kernel_launch(void* const* d_in, const int* in_sizes, int n_in,
                              void* d_out, int out_size, void* d_ws, size_t ws_size,
                              hipStream_t stream) {
  (void)in_sizes; (void)n_in; (void)out_size; (void)ws_size;
  const float* proposals = (const float*)d_in[0];
  const float* gt_boxes  = (const float*)d_in[1];
  const int*   gt_labels = (const int*)d_in[2];
  const float* logits    = (const float*)d_in[3];
  const float* regr      = (const float*)d_in[4];
  float* out = (float*)d_out;

  char* w = (char*)d_ws;
  size_t off = 0;
  auto alloc = [&](size_t bytes) -> void* {
    off = (off + 255) & ~(size_t)255;
    void* p = w + off;
    off += bytes;
    return p;
  };
  float*              sm       = (float*)alloc((size_t)B_ * NC_ * 4);
  unsigned*           hist     = (unsigned*)alloc((size_t)B_ * NBUCKET_ * 4);
  unsigned*           selA_cnt = (unsigned*)alloc((size_t)B_ * 4);
  unsigned*           selB_cnt = (unsigned*)alloc((size_t)B_ * 4);
  int*                tstar    = (int*)alloc((size_t)B_ * 4);
  unsigned long long* selA     = (unsigned long long*)alloc((size_t)B_ * TOPK_ * 8);
  unsigned long long* selB     = (unsigned long long*)alloc((size_t)B_ * BCAP_ * 8);
  float*              cb       = (float*)alloc((size_t)B_ * TOPK_ * 4 * 4);
  float*              ob       = (float*)alloc((size_t)B_ * TOPK_ * 4 * 4);
  int*                cl       = (int*)alloc((size_t)B_ * TOPK_ * 4);
  float*              cs       = (float*)alloc((size_t)B_ * TOPK_ * 4);
  unsigned*           keepInit = (unsigned*)alloc((size_t)B_ * 64 * 4);
  unsigned*           maskbuf  = (unsigned*)alloc((size_t)B_ * TOPK_ * 64 * 4);

  k0_init<<<(B_ * NBUCKET_ + 255) / 256, 256, 0, stream>>>(hist, selA_cnt, selB_cnt,
                                                           keepInit, out);
  k1_match<<<dim3(B_, (NP_ + 255) / 256), 256, 0, stream>>>(proposals, gt_boxes,
                                                            gt_labels, out);
  k2_scores<<<dim3(B_, (N_ / 16 + 7) / 8), 256, 0, stream>>>(proposals, logits, regr,
                                                             sm, hist);
  k3_thresh<<<1, 64, 0, stream>>>(hist, tstar);
  k4_compact<<<dim3(B_, (NC_ + 255) / 256), 256, 0, stream>>>(sm, tstar, selA_cnt,
                                                              selB_cnt, selA, selB);
  k5_sort<<<B_, 1024, 0, stream>>>(selA_cnt, selB_cnt, selA, selB, proposals, regr,
                                   cb, ob, cl, cs, keepInit);
  k6_mask<<<dim3(B_, TOPK_ / 8), 256, 0, stream>>>(ob, maskbuf);
  k7_scan<<<B_, 32, 0, stream>>>(maskbuf, keepInit, cb, cl, cs, out);
}